// De_conv_batched_multimasks_56521769615511
// MI455X (gfx1250) — compile-verified
//
#include <hip/hip_runtime.h>
#include <hip/hip_bf16.h>

#define INV_2PI 0.15915494309189535f

// v_sin_f32 computes sin(2*pi * x) (hardware input is pre-normalized by 1/2pi),
// which is exactly the reference's sin(TWO_PI * x).
__device__ __forceinline__ float diff_round(float x) {
    return __builtin_fmaf(-INV_2PI, __builtin_amdgcn_sinf(x), x);
}
__device__ __forceinline__ float harder_dr(float x) {
    return diff_round(diff_round(x));
}
// eq with the mid operand pre-rounded (hb = harder_dr(mid_c), hoisted per wave)
__device__ __forceinline__ float deq_pre(float a, float hb) {
    a = harder_dr(a);
    return harder_dr(__builtin_fmaf(a, hb, (1.0f - a) * (1.0f - hb)));
}
__device__ __forceinline__ float dand(float a, float b) {
    return diff_round(a) * diff_round(b);
}

__device__ __forceinline__ float mc_pixel(const float4 m, float hm0, float hm1,
                                          float hm2, float hm3) {
    return dand(dand(deq_pre(m.x, hm0), deq_pre(m.y, hm1)),
                dand(deq_pre(m.z, hm2), deq_pre(m.w, hm3)));
}

// faithful erosion incl. the reference's roll/zeroing quirks (a_u wraps at h==0)
__device__ __forceinline__ float eroded_times_edge(const float* __restrict__ row,
                                                   int p, float a, float ev) {
    const int w = p >> 3, h = p & 7;
    float ab = (w == 7) ? 0.0f : row[p + 8];
    float af = (w == 0) ? 0.0f : row[p - 8];
    float sh = row[(w << 3) | ((h + 7) & 7)];   // a[w, (h-1) mod 8]
    float au = (h == 7) ? 0.0f : sh;
    float ad = (h == 0) ? 0.0f : sh;
    float ed   = 1.0f - (ab * af + (1.0f - ab) * (1.0f - af));
    float edud = 1.0f - (au * ad + (1.0f - au) * (1.0f - ad));
    float s0a = ed * a;
    float s0b = edud * a;
    // differentiable_or_simple(s0a, s0b) = a*b + (1-a)*a + (1-b)*a
    float should0 = s0a * s0b + (1.0f - s0a) * s0a + (1.0f - s0b) * s0a;
    return (1.0f - should0) * a * ev;
}

__global__ __launch_bounds__(256) void
deconv_area_loss_kernel(const float* __restrict__ img,   // [P,8,8,1]
                        const float* __restrict__ mask,  // [P,8,8,4]
                        const float* __restrict__ mids,  // [P,4]
                        const float* __restrict__ edge,  // [P,8,8]
                        float* __restrict__ out,         // [P]
                        int nprob) {
    __shared__ __align__(16) float smask[8][256]; // 1 KB mask tile per wave
    __shared__ __align__(8)  float smc[8][64];    // mc tile per wave

    const int lane = threadIdx.x & 31;
    const int wid  = __builtin_amdgcn_readfirstlane((int)(threadIdx.x >> 5));
    const int prob = blockIdx.x * 8 + wid;
    if (prob >= nprob) return;   // whole-wave guard: EXEC stays full below

    // lane owns pixels p0 = 2*lane, p1 = 2*lane+1 (contiguous 32B of mask, 8B of img/edge)
    const int p0 = lane << 1;

    // ---- async DMA: 1 KB mask tile -> LDS (ASYNCcnt engine, CDNA5) ----
    const float* mbase = mask + (size_t)prob * 256;
    unsigned lds0 = (unsigned)(size_t)(&smask[wid][0]) + (unsigned)(lane * 32);
    int voff = lane * 32;
    asm volatile("global_load_async_to_lds_b128 %0, %1, %2"
                 :: "v"(lds0), "v"(voff), "s"(mbase) : "memory");
    asm volatile("global_load_async_to_lds_b128 %0, %1, %2 offset:16"
                 :: "v"(lds0 + 16u), "v"(voff), "s"(mbase) : "memory");

    // ---- overlap DMA with: uniform mid loads + img/edge b64 loads ----
    const float* midp = mids + (size_t)prob * 4;
    const float mid0 = midp[0], mid1 = midp[1], mid2 = midp[2], mid3 = midp[3];
    const float2 iv = ((const float2*)(img  + (size_t)prob * 64))[lane];
    const float2 ev = ((const float2*)(edge + (size_t)prob * 64))[lane];

    // hoist the wave-uniform harder_diff_round(mid_c): 8 sins instead of 16
    const float hm0 = harder_dr(mid0);
    const float hm1 = harder_dr(mid1);
    const float hm2 = harder_dr(mid2);
    const float hm3 = harder_dr(mid3);

    asm volatile("s_wait_asynccnt 0" ::: "memory");

    // ---- mc for both pixels (22 v_sin_f32 each), shared via LDS for erosion ----
    const float4* mp = (const float4*)(&smask[wid][0]);
    const float mc0 = mc_pixel(mp[p0],     hm0, hm1, hm2, hm3);
    const float mc1 = mc_pixel(mp[p0 + 1], hm0, hm1, hm2, hm3);
    *(float2*)(&smc[wid][p0]) = make_float2(mc0, mc1);   // one ds_store_b64
    // DS ops from one wave execute in order; fence only blocks compiler motion.
    asm volatile("" ::: "memory");

    const float* row = &smc[wid][0];
    float s_me = eroded_times_edge(row, p0,     mc0, ev.x)
               + eroded_times_edge(row, p0 + 1, mc1, ev.y);

    const float mi0 = mc0 * iv.x;
    const float mi1 = mc1 * iv.y;
    float s_mc = mc0 + mc1;
    float s_mi = mi0 + mi1;

    #pragma unroll
    for (int off = 16; off > 0; off >>= 1) {
        s_mc += __shfl_xor(s_mc, off, 32);
        s_mi += __shfl_xor(s_mi, off, 32);
        s_me += __shfl_xor(s_me, off, 32);
    }

    const float denom = s_mc + 1e-8f;
    const float meann = s_mi / denom;
    const float t0 = (mi0 - meann) * mc0;
    const float t1 = (mi1 - meann) * mc1;
    float v = t0 * t0 + t1 * t1;
    #pragma unroll
    for (int off = 16; off > 0; off >>= 1) v += __shfl_xor(v, off, 32);

    if (lane == 0) {
        const float loss = s_me * (1.0f / 64.0f);
        out[prob] = (v / denom) * loss * 1000.0f;
    }
}

extern "C" void kernel_launch(void* const* d_in, const int* in_sizes, int n_in,
                              void* d_out, int out_size, void* d_ws, size_t ws_size,
                              hipStream_t stream) {
    const float* img  = (const float*)d_in[0];  // resized_image [B,N,8,8,1]
    const float* mask = (const float*)d_in[1];  // mask_combined [B,N,8,8,4]
    const float* mids = (const float*)d_in[2];  // initial_masks [B,N,4]
    const float* edge = (const float*)d_in[3];  // edge_map      [B,N,8,8]
    float* out = (float*)d_out;                 // [B,N] float32

    const int nprob  = in_sizes[2] / 4;         // B*N
    const int blocks = (nprob + 7) / 8;         // 8 waves (problems) per block
    deconv_area_loss_kernel<<<blocks, 256, 0, stream>>>(img, mask, mids, edge, out, nprob);
}